// VectorQuantizer_38869454029580
// MI455X (gfx1250) — compile-verified
//
#include <hip/hip_runtime.h>
#include <hip/hip_bf16.h>
#include <math.h>

// ---------------------------------------------------------------------------
// VQ-VAE vector quantizer for MI455X (gfx1250, wave32)
//   z   : [32,32,64,64] f32  -> zf [65536, 64]
//   emb : [1024, 64] f32
// Outputs (concat, f32): loss(1) | z_q_st(4194304) | perplexity(1) | idx(65536)
// Main GEMM (65536x1024x64): V_WMMA_F32_16X16X4_F32 (full fp32 matrix path).
// Codebook staging to LDS: TENSOR_LOAD_TO_LDS (TDM) with hardware padding
// reproducing the 68-float conflict-free pitch.
// ---------------------------------------------------------------------------

typedef float v2f __attribute__((ext_vector_type(2)));
typedef float v8f __attribute__((ext_vector_type(8)));
typedef unsigned int u32x4 __attribute__((ext_vector_type(4)));
typedef int i32x8 __attribute__((ext_vector_type(8)));
typedef int i32x4 __attribute__((ext_vector_type(4)));

#define N_TOK   65536      // 32*32*64 rows
#define E_DIM   64
#define N_E     1024
#define LDSPITCH 68        // padded row stride (floats): conflict-free v2f reads

// d_out offsets (floats)
#define OFF_ZQ   1
#define OFF_PERP 4194305   // 1 + N_TOK*E_DIM
#define OFF_IDX  4194306

// d_ws layout (floats):
//  [0] tril_sum  [1] mse_sum  [2..1025] en  [1026..2049] counts
//  [2050 ...]    idx (as int) [N_TOK]

__device__ __forceinline__ v8f wmma_f32_k4(v2f a, v2f b, v8f c) {
  // D = A(16x4,f32) * B(4x16,f32) + C(16x16,f32)  -- full precision matrix op
  return __builtin_amdgcn_wmma_f32_16x16x4_f32(
      /*neg_a=*/false, a, /*neg_b=*/false, b,
      /*c_mod=*/(short)0, c, /*reuse_a=*/false, /*reuse_b=*/false);
}

// ---------------------------------------------------------------- init ws ---
__global__ void k_init(float* __restrict__ wf) {
  int t = blockIdx.x * 256 + threadIdx.x;
  if (t < 2050) wf[t] = 0.0f;
}

// ------------------------------------------------------- en[j] = ||e_j||^2 --
__global__ void k_en(const float* __restrict__ emb, float* __restrict__ en) {
  int j = blockIdx.x * 256 + threadIdx.x;
  if (j >= N_E) return;
  const float4* r = (const float4*)(emb + (size_t)j * E_DIM);
  float s = 0.0f;
#pragma unroll
  for (int c = 0; c < E_DIM / 4; ++c) {
    float4 v = r[c];
    s += v.x * v.x + v.y * v.y + v.z * v.z + v.w * v.w;
  }
  en[j] = s;
}

// --------------------------------- sum of tril(sqrt(max(d2,0))) over codes --
__global__ __launch_bounds__(256)
void k_tril(const float* __restrict__ emb, const float* __restrict__ en,
            float* __restrict__ tril_acc) {
  if (blockIdx.x > blockIdx.y) return;  // whole tile is strictly upper -> zero
  __shared__ float sA[16][LDSPITCH];
  __shared__ float sB[16][LDSPITCH];
  __shared__ float red[256];
  const int tx = threadIdx.x, ty = threadIdx.y;
  const int i = blockIdx.y * 16 + ty;   // row
  const int j = blockIdx.x * 16 + tx;   // col
  // stage 16+16 emb rows (each thread: one float4 of each)
  {
    const float4* a4 = (const float4*)(emb + (size_t)i * E_DIM);
    const float4* b4 = (const float4*)(emb + (size_t)(blockIdx.x * 16 + ty) * E_DIM);
    float4 va = a4[tx];
    float4 vb = b4[tx];
    *(float4*)&sA[ty][tx * 4] = va;
    *(float4*)&sB[ty][tx * 4] = vb;
  }
  __syncthreads();
  float val = 0.0f;
  if (i >= j) {
    float dot = 0.0f;
#pragma unroll
    for (int k = 0; k < E_DIM; ++k) dot = fmaf(sA[ty][k], sB[tx][k], dot);
    float d2 = en[i] + en[j] - 2.0f * dot;
    val = sqrtf(fmaxf(d2, 0.0f));
  }
  const int t = ty * 16 + tx;
  red[t] = val;
  __syncthreads();
  for (int s = 128; s > 0; s >>= 1) {
    if (t < s) red[t] += red[t + s];
    __syncthreads();
  }
  if (t == 0) atomicAdd(tril_acc, red[0]);
}

// ------------------------------------------- main: WMMA GEMM + row argmin ---
__global__ __launch_bounds__(256)
void k_vq(const float* __restrict__ z, const float* __restrict__ emb,
          const float* __restrict__ en, int* __restrict__ idx_out,
          float* __restrict__ counts, float* __restrict__ out_idx_f) {
  extern __shared__ float lds[];            // emb padded [1024*68] + en [1024]
  float* enS = lds + N_E * LDSPITCH;

  // --- Stage full codebook into LDS via the Tensor Data Mover (one wave). ---
  // D# describes a 1024x64 f32 tile; hardware padding (4 DWORDs after every
  // 64 DWORDs) produces the 68-float pitch the WMMA loop reads conflict-free.
  if (threadIdx.x < 32) {
    const unsigned long long ga = (unsigned long long)emb;
    u32x4 g0;
    g0[0] = 1u;                                   // count=1 (valid descriptor)
    g0[1] = 0u;                                   // lds_addr = 0 (base of lds[])
    g0[2] = (unsigned int)ga;                     // global_addr[31:0]
    g0[3] = (unsigned int)(ga >> 32) | (2u << 30);// global_addr[56:32] | type=2
    i32x8 g1;
    g1[0] = (int)((2u << 16) |                    // data_size = 4 bytes
                  (1u << 20) |                    // pad_enable
                  (5u << 22) |                    // pad_interval: 64 DWORDs
                  (3u << 25));                    // pad_amount:   4 DWORDs
    g1[1] = (int)(64u << 16);                     // tensor_dim0 = 64 (lo16)
    g1[2] = (int)(1024u << 16);                   // tensor_dim1 = 1024 (lo16)
    g1[3] = (int)(64u << 16);                     // tile_dim0 = 64
    g1[4] = 1024;                                 // tile_dim1 = 1024
    g1[5] = 64;                                   // tensor_dim0_stride = 64
    g1[6] = 0;
    g1[7] = 0;
    i32x4 g2 = {0, 0, 0, 0};
    i32x4 g3 = {0, 0, 0, 0};
    i32x8 g4 = {0, 0, 0, 0, 0, 0, 0, 0};          // extra group (6-arg form)
    __builtin_amdgcn_tensor_load_to_lds(g0, g1, g2, g3, g4, /*cpol=*/0);
    __builtin_amdgcn_s_wait_tensorcnt(0);
  }
  // Concurrently stage en[] on the normal path.
  for (int r = threadIdx.x; r < N_E; r += 256) enS[r] = en[r];
  __syncthreads();

  const int wave = threadIdx.x >> 5;
  const int lane = threadIdx.x & 31;
  const int half = lane >> 4;    // K sub-pair select (f32 16x16x4 layout)
  const int m    = lane & 15;    // row (A) / col (B,C) index
  const v2f* z2 = (const v2f*)z;

  for (int rt = blockIdx.x * 8 + wave; rt < N_TOK / 16; rt += gridDim.x * 8) {
    const int row0 = rt << 4;
    // A fragments: 16 rows x K=64, resident in VGPRs, reused for all 64 tiles
    v2f a[16];
    const v2f* arow = z2 + (size_t)(row0 + m) * (E_DIM / 2) + half;
#pragma unroll
    for (int kk = 0; kk < 16; ++kk) a[kk] = arow[kk * 2];

    float best[8];
    int bidx[8];
#pragma unroll
    for (int v = 0; v < 8; ++v) { best[v] = 3.4e38f; bidx[v] = 0; }

    for (int jt = 0; jt < N_E / 16; ++jt) {
      const int j = (jt << 4) + m;  // this lane's column in the C tile
      const v2f* brow = (const v2f*)(lds + (size_t)j * LDSPITCH) + half;
      v8f c0 = {0.f, 0.f, 0.f, 0.f, 0.f, 0.f, 0.f, 0.f};
      v8f c1 = {0.f, 0.f, 0.f, 0.f, 0.f, 0.f, 0.f, 0.f};
#pragma unroll
      for (int kk = 0; kk < 16; kk += 2) {   // two chains -> pipelined WMMAs
        c0 = wmma_f32_k4(a[kk],     brow[kk * 2],     c0);
        c1 = wmma_f32_k4(a[kk + 1], brow[kk * 2 + 2], c1);
      }
      const float enj = enS[j];
#pragma unroll
      for (int v = 0; v < 8; ++v) {
        float s = enj - 2.0f * (c0[v] + c1[v]);   // ||e||^2 - 2 z.e
        if (s < best[v]) { best[v] = s; bidx[v] = j; }  // strict <: first min
      }
    }

    // reduce argmin across the 16 lanes sharing each row (xor masks stay in half)
#pragma unroll
    for (int v = 0; v < 8; ++v) {
#pragma unroll
      for (int off = 8; off > 0; off >>= 1) {
        float ov = __shfl_xor(best[v], off, 32);
        int oi = __shfl_xor(bidx[v], off, 32);
        if (ov < best[v] || (ov == best[v] && oi < bidx[v])) {
          best[v] = ov;
          bidx[v] = oi;
        }
      }
      if (m == v) {  // lanes v and 16+v own rows row0+v and row0+8+v
        int row = row0 + v + (half << 3);
        idx_out[row] = bidx[v];
        out_idx_f[row] = (float)bidx[v];
        atomicAdd(&counts[bidx[v]], 1.0f);
      }
    }
  }
}

// ------------------------------------- gather z_q + MSE partial reduction ---
__global__ __launch_bounds__(256)
void k_gather(const float* __restrict__ z, const float* __restrict__ emb,
              const int* __restrict__ idx, float* __restrict__ out_zq,
              float* __restrict__ mse_acc) {
  __shared__ float red[256];
  const int tid = blockIdx.x * 256 + threadIdx.x;
  const int row = tid >> 6, e = tid & 63;
  const int j = idx[row];
  const float q = emb[((size_t)j << 6) + e];
  out_zq[tid] = q;                 // z_q_st == z + (z_q - z) == z_q numerically
  const float d = q - z[tid];
  red[threadIdx.x] = d * d;
  __syncthreads();
  for (int s = 128; s > 0; s >>= 1) {
    if (threadIdx.x < s) red[threadIdx.x] += red[threadIdx.x + s];
    __syncthreads();
  }
  if (threadIdx.x == 0) atomicAdd(mse_acc, red[0]);
}

// --------------------------------------------- loss + perplexity finalize ---
__global__ void k_final(const float* __restrict__ wf, float* __restrict__ out) {
  __shared__ float red[1024];
  const int t = threadIdx.x;
  const float c = wf[1026 + t];                 // counts[t]
  const float em = c * (1.0f / (float)N_TOK);   // e_mean
  red[t] = em * logf(em + 1e-10f);
  __syncthreads();
  for (int s = 512; s > 0; s >>= 1) {
    if (t < s) red[t] += red[t + s];
    __syncthreads();
  }
  if (t == 0) {
    out[OFF_PERP] = expf(-red[0]);
    const float mse = wf[1] * (1.0f / (float)(N_TOK * E_DIM));
    const float tril_mean = wf[0] * (1.0f / ((float)N_E * (float)N_E));
    // loss = mean((zq-z)^2) + 0.25*mean((zq-z)^2) + exp(-tril_mean/0.1)
    out[0] = 1.25f * mse + expf(-tril_mean * 10.0f);
  }
}

// ---------------------------------------------------------------------------
extern "C" void kernel_launch(void* const* d_in, const int* in_sizes, int n_in,
                              void* d_out, int out_size, void* d_ws, size_t ws_size,
                              hipStream_t stream) {
  const float* z   = (const float*)d_in[0];
  const float* emb = (const float*)d_in[1];
  float* out = (float*)d_out;

  float* wf     = (float*)d_ws;
  float* en     = wf + 2;
  float* counts = wf + 1026;
  int*   idx    = (int*)(wf + 2050);

  k_init<<<9, 256, 0, stream>>>(wf);
  k_en<<<4, 256, 0, stream>>>(emb, en);
  k_tril<<<dim3(64, 64), dim3(16, 16), 0, stream>>>(emb, en, wf /*tril at [0]*/);

  const size_t lds_bytes = (size_t)(N_E * LDSPITCH + N_E) * sizeof(float);
  k_vq<<<512, 256, lds_bytes, stream>>>(z, emb, en, idx, counts, out + OFF_IDX);

  k_gather<<<N_TOK * E_DIM / 256, 256, 0, stream>>>(z, emb, idx, out + OFF_ZQ,
                                                    wf + 1 /*mse*/);
  k_final<<<1, 1024, 0, stream>>>(wf, out);
}